// RoIHeads_fpn_35416300323750
// MI455X (gfx1250) — compile-verified
//
#include <hip/hip_runtime.h>
#include <hip/hip_bf16.h>
#include <stdint.h>

// Problem constants (fixed by the reference)
#define BIMG   16
#define NPROP  2048
#define NCLS   3
#define KDET   100
#define TPB    256
#define NEGV   (-1.0e9f)

typedef unsigned int u32;
typedef u32 v4u_t __attribute__((ext_vector_type(4)));
typedef int v8i_t __attribute__((ext_vector_type(8)));
typedef int v4i_t __attribute__((ext_vector_type(4)));

// ---- LDS layout (in floats) ----
#define OFF_LOGIT 0                       // 2048*3
#define OFF_DELTA (NPROP * 3)             // 2048*2 (class-c delta pair, TDM-strided)
#define OFF_PROP  (NPROP * 5)             // 2048*2
#define OFF_SCORE (NPROP * 7)             // 2048
#define OFF_LO    (NPROP * 8)             // 2048
#define OFF_HI    (NPROP * 9)             // 2048
#define OFF_MASK  (NPROP * 10)            // 64 u32 words (keep bitmask)
#define SMEM_FLOATS (NPROP * 10 + 64)
#define SMEM_BYTES  (SMEM_FLOATS * 4)

// ---------------------------------------------------------------------------
// Tensor Data Mover: DMA a (rows x row_elems) f32 tile (row pitch = pitch
// elems) from global to LDS. rows==0 => 1-D transfer of row_elems elements.
// Descriptor bit layout per CDNA5 ISA section 8 (D# groups 0/1).
// ---------------------------------------------------------------------------
__device__ inline void tdm_load_f32(u32 lds_off, const void* g,
                                    u32 row_elems, u32 rows, u32 pitch) {
#if __has_builtin(__builtin_amdgcn_tensor_load_to_lds)
  unsigned long long ga = (unsigned long long)(uintptr_t)g;
  v4u_t g0 = {
      1u,                                             // count=1, user mode
      lds_off,                                        // lds_addr (bytes)
      (u32)(ga & 0xFFFFFFFFu),                        // global_addr[31:0]
      ((u32)((ga >> 32) & 0x01FFFFFFu)) | (2u << 30)  // addr[56:32] | type=2
  };
  v8i_t g1 = {
      (int)(2u << 16),                                          // data_size=4B
      (int)((row_elems & 0xFFFFu) << 16),                       // tensor_dim0 lo16
      (int)(((row_elems >> 16) & 0xFFFFu) | ((rows & 0xFFFFu) << 16)), // dim0 hi | dim1 lo
      (int)(((rows >> 16) & 0xFFFFu) | ((row_elems & 0xFFFFu) << 16)), // dim1 hi | tile_dim0
      (int)(rows & 0xFFFFu),                                    // tile_dim1 | tile_dim2=0
      (int)pitch,                                               // tensor_dim0_stride lo32
      0, 0};
  v4i_t z4 = {0, 0, 0, 0};
#if defined(__clang_major__) && (__clang_major__ >= 23)
  v8i_t z8 = {0, 0, 0, 0, 0, 0, 0, 0};
  __builtin_amdgcn_tensor_load_to_lds(g0, g1, z4, z4, z8, 0);
#else
  __builtin_amdgcn_tensor_load_to_lds(g0, g1, z4, z4, 0);
#endif
#else
  (void)lds_off; (void)g; (void)row_elems; (void)rows; (void)pitch;
#endif
}

__device__ inline void wait_tensorcnt0() {
#if __has_builtin(__builtin_amdgcn_s_wait_tensorcnt)
  __builtin_amdgcn_s_wait_tensorcnt(0);
#else
  asm volatile("s_wait_tensorcnt 0" ::: "memory");
#endif
}

// ---------------------------------------------------------------------------
// One workgroup per (image b, fg class c). 256 threads.
// ---------------------------------------------------------------------------
__global__ __launch_bounds__(TPB)
void roi_post_kernel(const float* __restrict__ logit,   // [B,N,3]
                     const float* __restrict__ reg,     // [B,N,6]
                     const float* __restrict__ prop,    // [B,N,2]
                     const int* __restrict__ imgp,      // [1]
                     float* __restrict__ out) {         // [B,2,K,3]
  extern __shared__ float smem[];
  float* s_logit = smem + OFF_LOGIT;
  float* s_delta = smem + OFF_DELTA;
  float* s_prop  = smem + OFF_PROP;
  float* s_score = smem + OFF_SCORE;
  float* s_lo    = smem + OFF_LO;
  float* s_hi    = smem + OFF_HI;
  u32*   s_mask  = (u32*)(smem + OFF_MASK);
  __shared__ int s_cur;

  const int tid = (int)threadIdx.x;
  const int bc  = (int)blockIdx.x;     // 0..31
  const int b   = bc >> 1;
  const int c   = bc & 1;              // fg class index; actual class = c+1
  const int outBase = bc * (KDET * 3);

  // ---- Stage 1: TDM the per-block slices into LDS -------------------------
#if __has_builtin(__builtin_amdgcn_tensor_load_to_lds)
  if ((tid >> 5) == 0) {  // wave 0 issues the DMAs (wave-level ops)
    tdm_load_f32((u32)(uintptr_t)(void*)s_logit,
                 logit + (size_t)b * NPROP * 3, NPROP * 3, 0, NPROP * 3);
    // Strided 2-D: pick the 2 deltas of class (c+1) out of each 6-wide row.
    tdm_load_f32((u32)(uintptr_t)(void*)s_delta,
                 reg + (size_t)b * NPROP * 6 + (size_t)(c + 1) * 2, 2, NPROP, 6);
    tdm_load_f32((u32)(uintptr_t)(void*)s_prop,
                 prop + (size_t)b * NPROP * 2, NPROP * 2, 0, NPROP * 2);
  }
  wait_tensorcnt0();
#else
  // Fallback: cooperative copies
  for (int t = tid; t < NPROP * 3; t += TPB) s_logit[t] = logit[(size_t)b * NPROP * 3 + t];
  for (int t = tid; t < NPROP; t += TPB) {
    s_delta[2 * t + 0] = reg[((size_t)b * NPROP + t) * 6 + (c + 1) * 2 + 0];
    s_delta[2 * t + 1] = reg[((size_t)b * NPROP + t) * 6 + (c + 1) * 2 + 1];
  }
  for (int t = tid; t < NPROP * 2; t += TPB) s_prop[t] = prop[(size_t)b * NPROP * 2 + t];
#endif
  __syncthreads();

  // ---- Stage 2: softmax + decode + clip + validity mask -------------------
  const float img = (float)(*imgp);
  for (int e = tid; e < NPROP; e += TPB) {
    float l0 = s_logit[3 * e + 0], l1 = s_logit[3 * e + 1], l2 = s_logit[3 * e + 2];
    float mx = fmaxf(l0, fmaxf(l1, l2));
    float e0 = __expf(l0 - mx), e1 = __expf(l1 - mx), e2 = __expf(l2 - mx);
    float sc = ((c == 0) ? e1 : e2) / (e0 + e1 + e2);

    float p0 = s_prop[2 * e], p1 = s_prop[2 * e + 1];
    float w   = p1 - p0;
    float ctr = p0 + 0.5f * w;
    float dx  = s_delta[2 * e + 0];                 // WX = 1
    float dw  = fminf(s_delta[2 * e + 1], 4.0f);    // WW = 1, clamp like torchvision
    float pc  = dx * w + ctr;
    float pw  = __expf(dw) * w;
    float lo  = pc - 0.5f * pw;
    float hi  = pc + 0.5f * pw;
    lo = fminf(fmaxf(lo, 0.0f), img);
    hi = fminf(fmaxf(hi, 0.0f), img);

    bool valid = ((hi - lo) >= 10.0f) && (sc >= 0.05f);
    s_score[e] = valid ? sc : NEGV;
    s_lo[e] = lo;
    s_hi[e] = hi;
  }
  __syncthreads();

  // ---- Stage 3: bitonic sort (descending score), payload (lo,hi) ----------
  for (int k = 2; k <= NPROP; k <<= 1) {
    for (int j = k >> 1; j > 0; j >>= 1) {
#pragma unroll
      for (int p = 0; p < (NPROP / 2) / TPB; ++p) {
        int t = tid + p * TPB;                       // 0..1023
        int i = ((t & ~(j - 1)) << 1) | (t & (j - 1));
        int x = i | j;
        bool up = (i & k) == 0;
        float a = s_score[i], q = s_score[x];
        bool sw = up ? (a < q) : (a > q);
        if (sw) {
          s_score[i] = q;  s_score[x] = a;
          float t0 = s_lo[i]; s_lo[i] = s_lo[x]; s_lo[x] = t0;
          float t1 = s_hi[i]; s_hi[i] = s_hi[x]; s_hi[x] = t1;
        }
      }
      __syncthreads();
    }
  }

  // ---- Stage 4: keep bitmask init (keep0 = s > 0.5*NEG) -------------------
  if (tid < (NPROP / 32)) {
    u32 m = 0;
    int eb = tid << 5;
#pragma unroll
    for (int q = 0; q < 32; ++q)
      if (s_score[eb + q] > 0.5f * NEGV) m |= (1u << q);
    s_mask[tid] = m;
  }
  __syncthreads();

  // ---- Stage 5: greedy NMS; emit top-K kept in sorted order ---------------
  int count = 0;
  int scanFrom = 0;
  while (count < KDET) {
    if (tid == 0) {
      int found = -1;
      if (scanFrom < NPROP) {
        int w = scanFrom >> 5;
        u32 m = s_mask[w] & (0xFFFFFFFFu << (scanFrom & 31));
        for (;;) {
          if (m) { found = (w << 5) + __builtin_ctz(m); break; }
          if (++w >= (NPROP / 32)) break;
          m = s_mask[w];
        }
      }
      s_cur = found;
    }
    __syncthreads();
    int cur = s_cur;
    if (cur < 0) break;

    float li = s_lo[cur];
    float hi2 = s_hi[cur];
    if (tid == 0) {
      float* o = &out[outBase + count * 3];
      o[0] = li; o[1] = hi2; o[2] = s_score[cur];
    }
    // Parallel suppression sweep over the remainder of the sorted list.
    float wi = hi2 - li;
    for (int e = cur + 1 + tid; e < NPROP; e += TPB) {
      u32 w = (u32)(e >> 5), bit = 1u << (e & 31);
      if (s_mask[w] & bit) {
        float el = s_lo[e], eh = s_hi[e];
        float inter = fmaxf(fminf(hi2, eh) - fmaxf(li, el), 0.0f);
        float uni = wi + (eh - el) - inter;
        if (inter > 0.5f * fmaxf(uni, 1e-8f)) atomicAnd(&s_mask[w], ~bit);
      }
    }
    ++count;
    scanFrom = cur + 1;
    __syncthreads();
  }

  // ---- Stage 6: fill remaining slots (stable top_k over NEG ties) ---------
  if (count < KDET && tid == 0) {
    int cc = count;
    for (int e = 0; e < NPROP && cc < KDET; ++e) {
      if (!(s_mask[e >> 5] & (1u << (e & 31)))) {   // non-kept, ascending pos
        float* o = &out[outBase + cc * 3];
        o[0] = s_lo[e]; o[1] = s_hi[e]; o[2] = NEGV;
        ++cc;
      }
    }
  }
}

extern "C" void kernel_launch(void* const* d_in, const int* in_sizes, int n_in,
                              void* d_out, int out_size, void* d_ws, size_t ws_size,
                              hipStream_t stream) {
  (void)in_sizes; (void)n_in; (void)out_size; (void)d_ws; (void)ws_size;
  const float* logit = (const float*)d_in[0];
  const float* reg   = (const float*)d_in[1];
  const float* prop  = (const float*)d_in[2];
  const int*   img   = (const int*)d_in[3];
  float* out = (float*)d_out;

  // ~82KB dynamic LDS per workgroup (gfx1250 WGP has 320KB).
  (void)hipFuncSetAttribute((const void*)roi_post_kernel,
                            hipFuncAttributeMaxDynamicSharedMemorySize,
                            SMEM_BYTES);

  dim3 grid(BIMG * (NCLS - 1));  // 32 (image, fg-class) pairs
  dim3 block(TPB);
  hipLaunchKernelGGL(roi_post_kernel, grid, block, SMEM_BYTES, stream,
                     logit, reg, prop, img, out);
}